// SLUGenNet_2345052144079
// MI455X (gfx1250) — compile-verified
//
#include <hip/hip_runtime.h>
#include <hip/hip_bf16.h>

typedef __attribute__((ext_vector_type(2))) float v2f;
typedef __attribute__((ext_vector_type(8))) float v8f;

#define GCN_F 32   // EMBSIZE (output feature width, two 16-wide WMMA tiles)

// ---------------- utility kernels ----------------

__global__ void k_zero(float* __restrict__ p, int n) {
    int i = blockIdx.x * blockDim.x + threadIdx.x;
    if (i < n) p[i] = 0.0f;
}

// deg[row] += 1 for every edge (row, col); edges include self-loops.
__global__ void k_deg(const int* __restrict__ edges, int E, float* __restrict__ deg) {
    int e = blockIdx.x * blockDim.x + threadIdx.x;
    if (e < E) {
        int r = edges[2 * e];
        atomicAdd(&deg[r], 1.0f);
    }
}

__global__ void k_rsqrt(float* __restrict__ d, int n) {
    int i = blockIdx.x * blockDim.x + threadIdx.x;
    if (i < n) d[i] = 1.0f / sqrtf(d[i]);
}

// agg[r][f] += zs[c][f] for each edge (r, c), f in [0,32)
__global__ void k_scatter(const int* __restrict__ edges, int E,
                          const float* __restrict__ zs, float* __restrict__ agg) {
    int t = blockIdx.x * blockDim.x + threadIdx.x;
    int e = t >> 5;
    int f = t & 31;
    if (e < E) {
        int r = edges[2 * e];
        int c = edges[2 * e + 1];
        atomicAdd(&agg[(size_t)r * GCN_F + f], zs[(size_t)c * GCN_F + f]);
    }
}

// out = relu(dinv[row] * agg); re-zero agg for reuse by the next layer.
__global__ void k_finalize(const float* __restrict__ dinv, float* __restrict__ agg,
                           float* __restrict__ out, int n) {
    int i = blockIdx.x * blockDim.x + threadIdx.x;
    if (i < n) {
        float v = dinv[i >> 5] * agg[i];
        out[i] = fmaxf(v, 0.0f);
        agg[i] = 0.0f;
    }
}

// ---------------- WMMA dense layer ----------------
// Computes zs[i][f] = dinv[i] * ( X[row_src(i)] @ W + bias )[f]   for f in [0,32)
// where row_src(i) = ids ? ids[i] : i  (fused embedding gather for layer 1).
//
// One wave per 16-row output tile; two f32 16x16 accumulators (cols 0-15 / 16-31);
// K-loop in steps of 4 using V_WMMA_F32_16X16X4_F32 (exact fp32, matches reference
// precision; throughput is irrelevant at 0.2 GFLOP total).
//
// A-frag (16x4 f32): lane L holds M = L%16; VGPR0 = K(2*(L/16)), VGPR1 = K(2*(L/16)+1).
// B-frag (4x16 f32): lane L holds N = L%16; VGPR0 = K(2*(L/16)), VGPR1 = K(+1).
// C/D: VGPR r holds (M = r + 8*(L/16), N = L%16).
__global__ __launch_bounds__(256)
void k_dense_wmma(const float* __restrict__ X, const int* __restrict__ ids, int K,
                  const float* __restrict__ W, const float* __restrict__ bias,
                  const float* __restrict__ dinv, float* __restrict__ out) {
    const int lane = threadIdx.x & 31;
    const int wave = threadIdx.x >> 5;
    const int tile = blockIdx.x * (blockDim.x >> 5) + wave;

    const int mrow = tile * 16 + (lane & 15);      // A-matrix row this lane feeds
    const int koff = 2 * (lane >> 4);              // K sub-offset within a 4-step
    const int col  = lane & 15;                    // B/N column this lane feeds

    const size_t src = ids ? (size_t)ids[mrow] : (size_t)mrow;
    const float* __restrict__ arow = X + src * (size_t)K;

    v8f c0 = {0.f, 0.f, 0.f, 0.f, 0.f, 0.f, 0.f, 0.f};
    v8f c1 = {0.f, 0.f, 0.f, 0.f, 0.f, 0.f, 0.f, 0.f};

    for (int k = 0; k < K; k += 4) {
        v2f a;
        a.x = arow[k + koff];
        a.y = arow[k + koff + 1];

        v2f b0, b1v;
        b0.x  = W[(size_t)(k + koff) * GCN_F + col];
        b0.y  = W[(size_t)(k + koff + 1) * GCN_F + col];
        b1v.x = W[(size_t)(k + koff) * GCN_F + col + 16];
        b1v.y = W[(size_t)(k + koff + 1) * GCN_F + col + 16];

        c0 = __builtin_amdgcn_wmma_f32_16x16x4_f32(false, a, false, b0,
                                                   (short)0, c0, false, false);
        c1 = __builtin_amdgcn_wmma_f32_16x16x4_f32(false, a, false, b1v,
                                                   (short)0, c1, false, false);
    }

    const int mbase = tile * 16 + 8 * (lane >> 4);
    for (int r = 0; r < 8; ++r) {
        const int m = mbase + r;
        const float s = dinv[m];
        out[(size_t)m * GCN_F + col]      = s * (c0[r] + bias[col]);
        out[(size_t)m * GCN_F + col + 16] = s * (c1[r] + bias[col + 16]);
    }
}

// ---------------- launch ----------------

extern "C" void kernel_launch(void* const* d_in, const int* in_sizes, int n_in,
                              void* d_out, int out_size, void* d_ws, size_t ws_size,
                              hipStream_t stream) {
    const int*   node_ids = (const int*)d_in[0];
    const int*   edges    = (const int*)d_in[1];   // [E,2] flattened, E = 2N-1
    const float* table    = (const float*)d_in[2]; // [V, EMBDIM]
    const float* W1       = (const float*)d_in[3]; // [EMBDIM, 32]
    const float* b1       = (const float*)d_in[4]; // [32]
    const float* W2       = (const float*)d_in[5]; // [32, 32]
    const float* b2       = (const float*)d_in[6]; // [32]
    float*       out      = (float*)d_out;         // [N, 32]

    const int N  = in_sizes[0];            // 12288 (multiple of 16)
    const int E  = in_sizes[1] / 2;        // 2N-1 edges
    const int K1 = in_sizes[3] / GCN_F;    // EMBDIM = 256

    // Workspace layout (floats): dinv[N] | agg[N*32] | zs[N*32] | h[N*32]
    float* dinv = (float*)d_ws;
    float* agg  = dinv + N;
    float* zs   = agg + (size_t)N * GCN_F;
    float* h    = zs  + (size_t)N * GCN_F;

    const int nzero  = N * (1 + GCN_F);    // dinv + agg are contiguous
    const int tiles  = N / 16;
    const int gblk   = tiles / 8;          // 8 waves (tiles) per 256-thread block
    const int nfeat  = N * GCN_F;

    // Degree + D^{-1/2}
    k_zero <<<(nzero + 255) / 256, 256, 0, stream>>>(dinv, nzero);
    k_deg  <<<(E + 255) / 256,     256, 0, stream>>>(edges, E, dinv);
    k_rsqrt<<<(N + 255) / 256,     256, 0, stream>>>(dinv, N);

    // Layer 1: zs = D^{-1/2} (emb @ W1 + b1); propagate; relu
    k_dense_wmma<<<gblk, 256, 0, stream>>>(table, node_ids, K1, W1, b1, dinv, zs);
    k_scatter <<<(E * GCN_F + 255) / 256, 256, 0, stream>>>(edges, E, zs, agg);
    k_finalize<<<(nfeat + 255) / 256,     256, 0, stream>>>(dinv, agg, h, nfeat);

    // Layer 2: zs = D^{-1/2} (h @ W2 + b2); propagate; relu -> d_out
    k_dense_wmma<<<gblk, 256, 0, stream>>>(h, nullptr, GCN_F, W2, b2, dinv, zs);
    k_scatter <<<(E * GCN_F + 255) / 256, 256, 0, stream>>>(edges, E, zs, agg);
    k_finalize<<<(nfeat + 255) / 256,     256, 0, stream>>>(dinv, agg, out, nfeat);
}